// MaxUnPooling2DArgMax_21646635172482
// MI455X (gfx1250) — compile-verified
//
#include <hip/hip_runtime.h>

// Max-unpooling scatter-add for MI455X (gfx1250).
// out[b, idx[b,e]] += in[b,e]; B=8, per-batch pooled elems = 2^20,
// per-batch output slab = 2^22 floats. Output (128 MiB) fits in the 192 MB L2,
// so we zero it with RT stores (L2-resident) and let the random-index
// global_atomic_add_f32 (no return) resolve in L2. Streamed inputs use
// non-temporal loads so they don't evict the output working set.
//
// (Scatter kernel placed first in the file so the disasm snippet shows its
// body: expect 3x global_load_b128 with NT hint + 4x global_atomic_add_f32.)

typedef __attribute__((ext_vector_type(4))) float f32x4;
typedef __attribute__((ext_vector_type(2))) long i64x2;

#define LOG2_PER_BATCH 20  // H*W*C = 128*128*64
#define LOG2_FLAT_OUT  22  // (2H)*(2W)*C = 256*256*64

__global__ __launch_bounds__(256) void unpool_scatter_kernel(
    const float* __restrict__ in, const long* __restrict__ idx,
    float* __restrict__ out, int n4) {
  int t = blockIdx.x * blockDim.x + threadIdx.x;
  if (t >= n4) return;

  // 4 consecutive elements per thread; batch stride (2^20) is a multiple of 4,
  // so all 4 share one batch.
  long e0 = (long)t << 2;
  long batch = e0 >> LOG2_PER_BATCH;
  float* ob = out + (batch << LOG2_FLAT_OUT);

  // Streamed once: non-temporal b128 loads (don't pollute L2 holding the output).
  f32x4 v = __builtin_nontemporal_load((const f32x4*)in + t);
  i64x2 i01 = __builtin_nontemporal_load((const i64x2*)idx + 2 * (long)t);
  i64x2 i23 = __builtin_nontemporal_load((const i64x2*)idx + 2 * (long)t + 1);

  // Non-returning hardware f32 atomics -> global_atomic_add_f32 (STOREcnt),
  // duplicates sum just like tf.scatter_nd / .at[].add.
  unsafeAtomicAdd(ob + i01.x, v.x);
  unsafeAtomicAdd(ob + i01.y, v.y);
  unsafeAtomicAdd(ob + i23.x, v.z);
  unsafeAtomicAdd(ob + i23.y, v.w);
}

__global__ __launch_bounds__(256) void unpool_zero_kernel(f32x4* __restrict__ out,
                                                          int n4) {
  int t = blockIdx.x * blockDim.x + threadIdx.x;
  if (t < n4) {
    f32x4 z = {0.0f, 0.0f, 0.0f, 0.0f};
    out[t] = z;  // global_store_b128, RT -> zeros land resident in L2
  }
}

extern "C" void kernel_launch(void* const* d_in, const int* in_sizes, int n_in,
                              void* d_out, int out_size, void* d_ws, size_t ws_size,
                              hipStream_t stream) {
  const float* in = (const float*)d_in[0];   // float32 pooled values
  const long* idx = (const long*)d_in[1];    // int64 flat argmax indices
  float* out = (float*)d_out;                // float32 (B, 2H, 2W, C)

  int n = in_sizes[0];       // 8,388,608 pooled elements
  int zn4 = out_size >> 2;   // 8,388,608 float4 stores to zero 128 MiB
  int n4 = n >> 2;           // 2,097,152 scatter threads (4 elems each)

  // 1) Zero the output; resident-in-L2 zeros make the atomics L2-local.
  unpool_zero_kernel<<<(zn4 + 255) / 256, 256, 0, stream>>>((f32x4*)out, zn4);
  // 2) Scatter-add (same stream -> ordered after the zeroing kernel).
  unpool_scatter_kernel<<<(n4 + 255) / 256, 256, 0, stream>>>(in, idx, out, n4);
}